// SSDLite_44633300140476
// MI455X (gfx1250) — compile-verified
//
#include <hip/hip_runtime.h>
#include <hip/hip_bf16.h>
#include <math.h>

#define NUM_CLASSES 21
#define PBOX 3000
#define BATCH 8
#define TOPK 200
#define CONF_T 0.01f
#define NMS_T 0.45f
#define SORT_N 4096

typedef unsigned int u32;
typedef unsigned long long u64;

// LDS layout (dynamic, manually aliased):
//  phase A (sort): u64 keys[4096]            @ 0       (32768 B)
//  phase B (nms) : float sbox[3000*4]        @ 0       (48000 B)   <- overwrites keys after regs hold them
//                  float sscore[3072]        @ 48000   (12288 B)
//                  u8    sAlive[3072]        @ 60288   ( 3072 B)
//                  int   kept[200]           @ 63360   (  800 B)
#define SBOX_OFF   0
#define SSCORE_OFF 48000
#define SALIVE_OFF 60288
#define KEPT_OFF   63360
#define SMEM_BYTES 64160

extern __shared__ __align__(16) unsigned char smem[];

__device__ __forceinline__ u32 ordF(float f) {
  u32 u = __float_as_uint(f);
  return (u & 0x80000000u) ? ~u : (u | 0x80000000u);
}
__device__ __forceinline__ float unordF(u32 o) {
  u32 u = (o & 0x80000000u) ? (o ^ 0x80000000u) : ~o;
  return __uint_as_float(u);
}

__global__ void ssd_zero_kernel(float* out, int n) {
  int i = blockIdx.x * blockDim.x + threadIdx.x;
  if (i < n) out[i] = 0.0f;
}

__global__ void ssd_decode_kernel(const float* __restrict__ loc,
                                  const float* __restrict__ priors,
                                  float* __restrict__ gbox) {
  int t = blockIdx.x * blockDim.x + threadIdx.x;
  if (t >= BATCH * PBOX) return;
  int p = t % PBOX;
  const float* l  = loc + (u64)t * 4;
  const float* pr = priors + (u64)p * 4;
  float cx = pr[0] + l[0] * 0.1f * pr[2];
  float cy = pr[1] + l[1] * 0.1f * pr[3];
  float w  = pr[2] * expf(l[2] * 0.2f);
  float h  = pr[3] * expf(l[3] * 0.2f);
  float x1 = cx - 0.5f * w, y1 = cy - 0.5f * h;
  float4 o = make_float4(x1, y1, x1 + w, y1 + h);
  *(float4*)(gbox + (u64)t * 4) = o;
}

__global__ __launch_bounds__(256) void ssd_nms_kernel(const float* __restrict__ conf,
                                                      const float* __restrict__ gbox,
                                                      float* __restrict__ out) {
  const int tid = threadIdx.x;
  const int cls = blockIdx.x + 1;   // skip background class 0
  const int b   = blockIdx.y;

  u64*   keys   = (u64*)smem;
  float* sbox   = (float*)(smem + SBOX_OFF);
  float* sscore = (float*)(smem + SSCORE_OFF);
  unsigned char* sAlive = smem + SALIVE_OFF;
  int*   kept   = (int*)(smem + KEPT_OFF);

  // ---- Phase 1: build sort keys (descending score, stable by index) ----
  for (int i = tid; i < SORT_N; i += 256) {
    u64 k;
    if (i < PBOX) {
      float s  = conf[(u64)(b * PBOX + i) * NUM_CLASSES + cls];
      float fs = (s > CONF_T) ? s : -__builtin_inff();
      k = ((u64)(~ordF(fs)) << 32) | (u32)i;
    } else {
      k = ~0ull;  // padding sorts last
    }
    keys[i] = k;
  }
  __syncthreads();

  // ---- Phase 2: bitonic sort ascending (4096 u64 keys in LDS) ----
  for (u32 kk = 2; kk <= SORT_N; kk <<= 1) {
    for (u32 j = kk >> 1; j > 0; j >>= 1) {
      for (u32 i = tid; i < SORT_N; i += 256) {
        u32 ixj = i ^ j;
        if (ixj > i) {
          u64 a = keys[i], c = keys[ixj];
          bool up = ((i & kk) == 0);
          if ((a > c) == up) { keys[i] = c; keys[ixj] = a; }
        }
      }
      __syncthreads();
    }
  }

  // ---- Phase 3: pull sorted entries into registers before aliasing LDS ----
  float rs[12];
  u32   ridx[12];
#pragma unroll
  for (int q = 0; q < 12; ++q) {
    int i = tid + q * 256;            // covers 0..3071; all valid entries live here
    u64 k = keys[i];
    rs[q]   = unordF(~(u32)(k >> 32));
    ridx[q] = (u32)k;
  }
  __syncthreads();                    // everyone done reading keys; safe to overwrite

  // ---- Phase 4: scatter scores/alive; async-gather boxes global->LDS (TDM-lite path) ----
#pragma unroll
  for (int q = 0; q < 12; ++q) {
    int i = tid + q * 256;
    float s = rs[q];
    bool valid = (s > CONF_T) && (ridx[q] < (u32)PBOX) && (i < PBOX);
    sscore[i] = s;
    sAlive[i] = valid ? 1 : 0;
    if (valid) {
      const float* gp = gbox + ((u64)b * PBOX + ridx[q]) * 4;
      u32 ldsOff = (u32)(u64)(void*)(sbox + (u64)i * 4);   // low 32 bits of generic ptr = LDS offset
      asm volatile("global_load_async_to_lds_b128 %0, %1, off"
                   :: "v"(ldsOff), "v"(gp) : "memory");
    }
  }
  asm volatile("s_wait_asynccnt 0x0" ::: "memory");
  __syncthreads();

  // ---- Phase 5: greedy NMS (sequential over rank, parallel suppression) ----
  int kc = 0;
  for (int i = 0; i < PBOX && kc < TOPK; ++i) {
    if (sAlive[i]) {                       // uniform: last write was before a barrier
      if (tid == 0) kept[kc] = i;
      float ix1 = sbox[4 * i + 0], iy1 = sbox[4 * i + 1];
      float ix2 = sbox[4 * i + 2], iy2 = sbox[4 * i + 3];
      float iarea = (ix2 - ix1) * (iy2 - iy1);
      for (int j = i + 1 + tid; j < PBOX; j += 256) {
        if (sAlive[j]) {
          float jx1 = sbox[4 * j + 0], jy1 = sbox[4 * j + 1];
          float jx2 = sbox[4 * j + 2], jy2 = sbox[4 * j + 3];
          float xx1 = fmaxf(ix1, jx1), yy1 = fmaxf(iy1, jy1);
          float xx2 = fminf(ix2, jx2), yy2 = fminf(iy2, jy2);
          float w = fmaxf(xx2 - xx1, 0.0f), h = fmaxf(yy2 - yy1, 0.0f);
          float inter = w * h;
          float jarea = (jx2 - jx1) * (jy2 - jy1);
          float uni = fmaxf(iarea + jarea - inter, 1e-12f);
          if (inter > NMS_T * uni) sAlive[j] = 0;   // iou > 0.45
        }
      }
      ++kc;
      __syncthreads();
    }
  }
  __syncthreads();

  // ---- Phase 6: emit [TOPK, 5] rows of (score, x1, y1, x2, y2) ----
  if (tid < TOPK) {
    float* o = out + (((u64)b * NUM_CLASSES + cls) * TOPK + tid) * 5;
    if (tid < kc) {
      int i = kept[tid];
      o[0] = sscore[i];
      o[1] = sbox[4 * i + 0];
      o[2] = sbox[4 * i + 1];
      o[3] = sbox[4 * i + 2];
      o[4] = sbox[4 * i + 3];
    } else {
      o[0] = 0.0f; o[1] = 0.0f; o[2] = 0.0f; o[3] = 0.0f; o[4] = 0.0f;
    }
  }
}

extern "C" void kernel_launch(void* const* d_in, const int* in_sizes, int n_in,
                              void* d_out, int out_size, void* d_ws, size_t ws_size,
                              hipStream_t stream) {
  (void)in_sizes; (void)n_in; (void)out_size; (void)ws_size;
  const float* loc    = (const float*)d_in[0];  // [8, 3000, 4]
  const float* conf   = (const float*)d_in[1];  // [8, 3000, 21]
  const float* priors = (const float*)d_in[2];  // [3000, 4]
  float* out  = (float*)d_out;                  // [8, 21, 200, 5]
  float* gbox = (float*)d_ws;                   // [8, 3000, 4] decoded boxes (384 KB)

  const int outN = BATCH * NUM_CLASSES * TOPK * 5;
  ssd_zero_kernel<<<(outN + 255) / 256, 256, 0, stream>>>(out, outN);
  ssd_decode_kernel<<<(BATCH * PBOX + 255) / 256, 256, 0, stream>>>(loc, priors, gbox);
  ssd_nms_kernel<<<dim3(NUM_CLASSES - 1, BATCH), 256, SMEM_BYTES, stream>>>(conf, gbox, out);
}